// FourierTimeActivation_9500467659384
// MI455X (gfx1250) — compile-verified
//
#include <hip/hip_runtime.h>

// Problem dims
#define Bn 4
#define Cn 16
#define Fn 35
#define Gn 72
#define Dn 24
#define Tn 256
#define NVOX (Dn * Dn * Dn) /* 13824 */

// Padded GEMM dims
#define FP 36   // K of GEMM1, 9 steps of 4
#define GP1 80  // M of GEMM1, 5 tiles of 16
#define FP2 48  // M of GEMM2, 3 tiles of 16
#define GK 72   // K of GEMM2, 18 steps of 4

typedef __attribute__((ext_vector_type(2))) float v2f;
typedef __attribute__((ext_vector_type(8))) float v8f;

// ---------------------------------------------------------------------------
// Kernel 1: per-(b,c) time bias: tb[b*C+c] = dot(t[b,:], W_time[c,:]) + b_time[c]
// ---------------------------------------------------------------------------
__global__ void time_bias_kernel(const float* __restrict__ t,
                                 const float* __restrict__ W,
                                 const float* __restrict__ bias,
                                 float* __restrict__ tb) {
  int i = threadIdx.x;          // 0..63
  if (i >= Bn * Cn) return;
  int b = i / Cn, c = i % Cn;
  float s = bias[c];
  for (int k = 0; k < Tn; ++k) s += t[b * Tn + k] * W[c * Tn + k];
  tb[i] = s;
}

// ---------------------------------------------------------------------------
// Kernel 2: fused  A_ft @ selu(A_ift @ x_hat + tb)  per (b,c) slab.
// 8 waves / block; each wave owns one 16-voxel column tile.
// ---------------------------------------------------------------------------
__launch_bounds__(256, 2)
__global__ void fourier_selu_kernel(const float* __restrict__ xh,
                                    const float* __restrict__ A_ift,
                                    const float* __restrict__ A_ft,
                                    const float* __restrict__ tb,
                                    float* __restrict__ out) {
  __shared__ float sA1[GP1 * FP];       // A_ift padded (80 x 36), zero fill
  __shared__ float sA2[FP2 * GK];       // A_ft  padded (48 x 72), zero fill
  __shared__ float sY[8 * Gn * 16];     // per-wave Y scratch (72 x 16 each)

  const int tid  = threadIdx.x;
  const int slab = blockIdx.x;          // b*C + c

  // Cooperative zero-padded load of the two small operand matrices.
  for (int i = tid; i < GP1 * FP; i += 256) {
    int g = i / FP, f = i % FP;
    sA1[i] = (g < Gn && f < Fn) ? A_ift[g * Fn + f] : 0.0f;
  }
  for (int i = tid; i < FP2 * GK; i += 256) {
    int f = i / GK, g = i % GK;
    sA2[i] = (f < Fn) ? A_ft[f * Gn + g] : 0.0f;
  }
  __syncthreads();

  const int lane = tid & 31;
  const int wave = tid >> 5;
  const int lh   = lane >> 4;           // lane-half: 0 = lanes 0-15, 1 = 16-31
  const int n    = lane & 15;           // column within the 16-wide tile

  const long base_n = (long)(blockIdx.y * 8 + wave) * 16;
  const float tbias = tb[slab];
  const float* xs = xh + (size_t)slab * Fn * NVOX + base_n + n;
  float* sYw = &sY[wave * Gn * 16];

  // ---------------- GEMM1: (80x36) @ (36x16) -> 5 accumulator tiles --------
  v8f acc1[5];
#pragma unroll
  for (int m = 0; m < 5; ++m)
#pragma unroll
    for (int j = 0; j < 8; ++j) acc1[m][j] = 0.0f;

#pragma unroll
  for (int k = 0; k < 9; ++k) {
    const int f0 = k * 4 + lh * 2;      // K rows held by this lane-half
    const int f1 = f0 + 1;
    v2f bf;
    bf.x = (f0 < Fn) ? xs[(size_t)f0 * NVOX] : 0.0f;
    bf.y = (f1 < Fn) ? xs[(size_t)f1 * NVOX] : 0.0f;
#pragma unroll
    for (int m = 0; m < 5; ++m) {
      const float* ap = &sA1[(m * 16 + n) * FP + f0];
      v2f af;
      af.x = ap[0];
      af.y = ap[1];
      acc1[m] = __builtin_amdgcn_wmma_f32_16x16x4_f32(
          false, af, false, bf, (short)0, acc1[m], false, false);
    }
  }

  // -------- bias + SELU in registers, re-layout rows->K via LDS ------------
  const float kScale = 1.0507009873554805f;
  const float kAlpha = 1.6732632423543772f;
#pragma unroll
  for (int m = 0; m < 5; ++m) {
#pragma unroll
    for (int j = 0; j < 8; ++j) {
      const int g = m * 16 + j + 8 * lh;   // C/D layout: vgpr j, lane-half lh
      if (g < Gn) {
        float v = acc1[m][j] + tbias;
        v = (v > 0.0f) ? kScale * v : kScale * kAlpha * (__expf(v) - 1.0f);
        sYw[g * 16 + n] = v;
      }
    }
  }
  // Same-wave LDS ops are in order (DScnt) — no barrier needed; scratch is
  // private to this wave.

  // ---------------- GEMM2: (48x72) @ (72x16) -> 3 accumulator tiles --------
  v8f acc2[3];
#pragma unroll
  for (int m = 0; m < 3; ++m)
#pragma unroll
    for (int j = 0; j < 8; ++j) acc2[m][j] = 0.0f;

#pragma unroll
  for (int k = 0; k < 18; ++k) {
    const int g0 = k * 4 + lh * 2;
    v2f bf;
    bf.x = sYw[g0 * 16 + n];
    bf.y = sYw[(g0 + 1) * 16 + n];
#pragma unroll
    for (int m = 0; m < 3; ++m) {
      const float* ap = &sA2[(m * 16 + n) * GK + g0];
      v2f af;
      af.x = ap[0];
      af.y = ap[1];
      acc2[m] = __builtin_amdgcn_wmma_f32_16x16x4_f32(
          false, af, false, bf, (short)0, acc2[m], false, false);
    }
  }

  // ---------------- store (rows f < 35 only) -------------------------------
  float* os = out + (size_t)slab * Fn * NVOX + base_n + n;
#pragma unroll
  for (int m = 0; m < 3; ++m) {
#pragma unroll
    for (int j = 0; j < 8; ++j) {
      const int f = m * 16 + j + 8 * lh;
      if (f < Fn) os[(size_t)f * NVOX] = acc2[m][j];
    }
  }
}

// ---------------------------------------------------------------------------
extern "C" void kernel_launch(void* const* d_in, const int* in_sizes, int n_in,
                              void* d_out, int out_size, void* d_ws, size_t ws_size,
                              hipStream_t stream) {
  const float* x_hat = (const float*)d_in[0];
  const float* t     = (const float*)d_in[1];
  const float* A_ift = (const float*)d_in[2];
  const float* A_ft  = (const float*)d_in[3];
  const float* W     = (const float*)d_in[4];
  const float* bvec  = (const float*)d_in[5];
  float* tb  = (float*)d_ws;            // 64 floats
  float* out = (float*)d_out;

  time_bias_kernel<<<1, 64, 0, stream>>>(t, W, bvec, tb);

  dim3 grid(Bn * Cn, NVOX / (16 * 8)); // 64 x 108
  fourier_selu_kernel<<<grid, 256, 0, stream>>>(x_hat, A_ift, A_ft, tb, out);
}